// Net_335007449248
// MI455X (gfx1250) — compile-verified
//
#include <hip/hip_runtime.h>

// Elementwise: out = clip(x * c[ch], 1e-8, 1) ^ (mask==0 ? g1[ch] : g2[ch])
// [B,C,H,W] = [16,3,512,512] fp32  ->  memory-bound streaming kernel.
// Roofline: 12 B/elem * 12.58M elem ~= 151 MB @ 23.3 TB/s => ~6.5 us; no matrix op exists
// in this workload, so the optimal CDNA5 path is b128 vmem + packed VALU + TRANS, not WMMA.

typedef float v4f __attribute__((ext_vector_type(4)));
typedef int   v4i __attribute__((ext_vector_type(4)));

#define HW       (512u * 512u)
#define HW4_LOG2 16u            // HW/4 = 65536 float4 vectors per (b,c) channel

__global__ __launch_bounds__(256) void net_pow_kernel(
    const v4f* __restrict__ img,   // low_img, viewed as float4
    const v4i* __restrict__ msk,   // I_Mask,  viewed as int4
    const float* __restrict__ g1,  // [B*C]
    const float* __restrict__ g2,  // [B*C]
    const float* __restrict__ cc,  // [B*C]
    v4f* __restrict__ out,
    unsigned nvec)
{
    unsigned i = blockIdx.x * 256u + threadIdx.x;
    if (i >= nvec) return;

    // Channel index is uniform across each wave32 (65536 % 32 == 0): force it
    // scalar so the 3 per-channel constants become s_load_b32 via the scalar
    // cache. Unsigned >> 16 keeps this a single s_lshr_b32 (no sign fixup).
    unsigned ch = (unsigned)__builtin_amdgcn_readfirstlane(i >> HW4_LOG2);
    float cs = cc[ch];
    float ga = g1[ch];
    float gb = g2[ch];

    // 128-bit streaming loads (RT policy: inputs stay L2-resident across replays)
    v4f x = img[i];
    v4i m = msk[i];

    v4f r;
#pragma unroll
    for (int k = 0; k < 4; ++k) {
        float v = x[k] * cs;                       // v_pk_mul_f32 pairs
        v = fminf(fmaxf(v, 1e-8f), 1.0f);          // clip -> v in (0, 1], pow is safe
        float g = (m[k] == 0) ? ga : gb;           // v_cndmask
        // v^g = exp2(g * log2(v)) via native TRANS ops (v_log_f32 / v_exp_f32)
        r[k] = __builtin_amdgcn_exp2f(g * __builtin_amdgcn_logf(v));
    }

    // Output is write-once / never re-read on device: non-temporal store keeps
    // the 50MB result from evicting the L2-resident inputs between replays.
    __builtin_nontemporal_store(r, &out[i]);
}

extern "C" void kernel_launch(void* const* d_in, const int* in_sizes, int n_in,
                              void* d_out, int out_size, void* d_ws, size_t ws_size,
                              hipStream_t stream) {
    // setup_inputs order: low_img, g1, g2, c, I_Mask
    const v4f*   img = (const v4f*)d_in[0];
    const float* g1  = (const float*)d_in[1];
    const float* g2  = (const float*)d_in[2];
    const float* cc  = (const float*)d_in[3];
    const v4i*   msk = (const v4i*)d_in[4];
    v4f*         out = (v4f*)d_out;

    unsigned nvec   = (unsigned)out_size / 4u;    // 16*3*512*512 / 4 = 3,145,728
    unsigned blocks = (nvec + 255u) / 256u;       // 12,288 blocks of 8 wave32s

    net_pow_kernel<<<blocks, 256, 0, stream>>>(img, msk, g1, g2, cc, out, nvec);
}